// ReduceSSDBoundingBoxes_43413529428595
// MI455X (gfx1250) — compile-verified
//
#include <hip/hip_runtime.h>
#include <hip/hip_bf16.h>

#define N_BOX   4774          // 60^2 + 30^2 + 15^2 + 7^2
#define NPAD    8192          // power-of-two padding for bitonic sort
#define SC_PAD  4776          // score slab padded so next slab is 16B aligned
#define PROB_T  0.9f
#define IOU_T   0.5f
#define WIDTH_F 320.0f
#define HEIGHT_F 240.0f
#define EPS_F   1e-9f

typedef __attribute__((ext_vector_type(4))) unsigned int v4u;
typedef __attribute__((ext_vector_type(8))) int          v8i;
typedef __attribute__((ext_vector_type(4))) int          v4i;

// ---------------------------------------------------------------------------
// Kernel 1: per-box transform -> rounded [x1,y1,x2,y2], score, sort key, index
// ---------------------------------------------------------------------------
__global__ void ssd_transform_kernel(const float* __restrict__ x,
                                     const float* __restrict__ mult,
                                     const float* __restrict__ priors,
                                     float* __restrict__ key,
                                     int*   __restrict__ sidx,
                                     float* __restrict__ bbx,
                                     float* __restrict__ score) {
  int i = blockIdx.x * blockDim.x + threadIdx.x;
  if (i >= NPAD) return;
  if (i < N_BOX) {
    __builtin_prefetch(&x[i * 5], 0, 1);   // global_prefetch_b8 on gfx1250
    float p  = x[i * 5 + 0];
    bool  m  = p > PROB_T;
    float bx = x[i * 5 + 1];
    float by = x[i * 5 + 2];
    float bw = x[i * 5 + 3];
    float bh = x[i * 5 + 4];
    if (m) {
      float mu = mult[i];
      bx = bx * mu + priors[i * 4 + 0];
      by = by * mu + priors[i * 4 + 1];
      bw = bw + priors[i * 4 + 2];
      bh = bh + priors[i * 4 + 3];
    }
    bx *= WIDTH_F;  by *= HEIGHT_F;
    bw *= WIDTH_F;  bh *= HEIGHT_F;
    float x1 = rintf(bx);          // round-half-even == jnp.round (RTE mode)
    float y1 = rintf(by);
    float x2 = rintf(bx + bw);
    float y2 = rintf(by + bh);
    bbx[i * 4 + 0] = x1;
    bbx[i * 4 + 1] = y1;
    bbx[i * 4 + 2] = x2;
    bbx[i * 4 + 3] = y2;
    score[i] = p;
    key[i]   = m ? p : -__builtin_inff();
    sidx[i]  = i;
  } else {
    key[i]  = -__builtin_inff();   // padding sorts after every real entry
    sidx[i] = i;                   // (index tiebreak: i >= N_BOX)
  }
}

// ---------------------------------------------------------------------------
// Kernel 2: single-workgroup bitonic sort (stable via index tiebreak) + gather
// ---------------------------------------------------------------------------
__global__ __launch_bounds__(1024)
void ssd_sort_kernel(const float* __restrict__ key,
                     const int*   __restrict__ sidx,
                     const float* __restrict__ bbx,
                     const float* __restrict__ score,
                     float* __restrict__ bbx_s,
                     float* __restrict__ sc_s) {
  __shared__ float sk[NPAD];
  __shared__ int   si[NPAD];
  const int tid = threadIdx.x;

  for (int i = tid; i < NPAD; i += 1024) { sk[i] = key[i]; si[i] = sidx[i]; }
  __syncthreads();

  for (unsigned k = 2; k <= NPAD; k <<= 1) {
    for (unsigned j = k >> 1; j > 0; j >>= 1) {
      for (unsigned i = tid; i < NPAD; i += 1024) {
        unsigned p = i ^ j;
        if (p > i) {
          float ka = sk[i], kb = sk[p];
          int   ia = si[i], ib = si[p];
          // "a before b": larger key first; equal keys -> smaller orig index
          bool before = (ka > kb) || ((ka == kb) && (ia < ib));
          bool up = ((i & k) == 0);
          if (up ? !before : before) {
            sk[i] = kb; sk[p] = ka;
            si[i] = ib; si[p] = ia;
          }
        }
      }
      __syncthreads();
    }
  }

  for (int t = tid; t < N_BOX; t += 1024) {
    int o = si[t];
    bbx_s[t * 4 + 0] = bbx[o * 4 + 0];
    bbx_s[t * 4 + 1] = bbx[o * 4 + 1];
    bbx_s[t * 4 + 2] = bbx[o * 4 + 2];
    bbx_s[t * 4 + 3] = bbx[o * 4 + 3];
    sc_s[t] = score[o];
  }
}

// ---------------------------------------------------------------------------
// Kernel 3: single-workgroup NMS. Boxes DMA'd into LDS by the Tensor Data
// Mover (TENSOR_LOAD_TO_LDS + s_wait_tensorcnt), then sequential-suppression
// scan with the whole O(n^2) IoU computed from LDS.
// ---------------------------------------------------------------------------
__global__ __launch_bounds__(1024)
void ssd_nms_kernel(const float* __restrict__ bbx_s,
                    const float* __restrict__ sc_s,
                    float* __restrict__ out,      // n*5 row-major
                    float* __restrict__ keep) {   // n
  __shared__ float4 box[N_BOX];                   // 76,384 B
  __shared__ float  area[N_BOX];                  // 19,096 B
  __shared__ unsigned char valid[N_BOX];
  __shared__ unsigned char supp[N_BOX];
  const int tid = threadIdx.x;

#if __has_builtin(__builtin_amdgcn_tensor_load_to_lds) && __has_builtin(__builtin_amdgcn_s_wait_tensorcnt)
  if ((tid >> 5) == 0) {   // wave 0 issues one TDM descriptor for the tile
    unsigned lds_off = (unsigned)(size_t)(&box[0]);       // LDS byte offset
    unsigned long long ga = (unsigned long long)(size_t)bbx_s;
    const unsigned elems = N_BOX * 4u;                    // f32 elements
    // D# group 0: count=1 | lds_addr | global_addr[56:0] | type=2
    v4u g0;
    g0[0] = 1u;
    g0[1] = lds_off;
    g0[2] = (unsigned)ga;
    g0[3] = (unsigned)((ga >> 32) & 0x01FFFFFFu) | (2u << 30);
    // D# group 1: data_size=4B, tensor_dim0=elems, tensor_dim1=1,
    //             tile_dim0=elems (1 row), tile_dim1=1, dim0_stride=elems
    v8i g1;
    g1[0] = (int)(2u << 16);                       // data_size = 4 bytes
    g1[1] = (int)((elems & 0xFFFFu) << 16);        // tensor_dim0[15:0]
    g1[2] = (int)(((elems >> 16) & 0xFFFFu) | (1u << 16)); // dim0 hi | dim1=1
    g1[3] = (int)((elems & 0xFFFFu) << 16);        // tile_dim0
    g1[4] = 1;                                     // tile_dim1=1, tile_dim2=0
    g1[5] = (int)elems;                            // tensor_dim0_stride lo
    g1[6] = 0;
    g1[7] = 0;
    v4i g2 = (v4i)0;                               // unused (<=2D tensor)
    v4i g3 = (v4i)0;                               // unused (<=2D tensor)
    v8i g4 = (v8i)0;                               // extra group, 6-arg form
    __builtin_amdgcn_tensor_load_to_lds(g0, g1, g2, g3, g4, 0);
    __builtin_amdgcn_s_wait_tensorcnt(0);
  }
#else
  for (int i = tid; i < N_BOX; i += 1024)
    box[i] = ((const float4*)bbx_s)[i];
#endif
  __syncthreads();

  for (int i = tid; i < N_BOX; i += 1024) {
    float4 b = box[i];
    area[i]  = (b.z - b.x) * (b.w - b.y);
    float s  = sc_s[i];
    valid[i] = (s > PROB_T) ? 1 : 0;
    supp[i]  = 0;
    out[i * 5 + 0] = s;
    out[i * 5 + 1] = b.x;
    out[i * 5 + 2] = b.y;
    out[i * 5 + 3] = b.z - b.x;
    out[i * 5 + 4] = b.w - b.y;
  }
  __syncthreads();

  for (int i = 0; i < N_BOX; ++i) {
    bool keep_i = valid[i] && !supp[i];
    if (tid == 0) keep[i] = keep_i ? 1.0f : 0.0f;
    if (keep_i) {
      float4 bi = box[i];
      float  ai = area[i];
      for (int j = i + 1 + tid; j < N_BOX; j += 1024) {
        float4 bj = box[j];
        float ltx = fmaxf(bi.x, bj.x);
        float lty = fmaxf(bi.y, bj.y);
        float rbx = fminf(bi.z, bj.z);
        float rby = fminf(bi.w, bj.w);
        float w = fmaxf(rbx - ltx, 0.0f);
        float h = fmaxf(rby - lty, 0.0f);
        float inter = w * h;
        float iou = inter / (ai + area[j] - inter + EPS_F);
        if (iou > IOU_T) supp[j] = 1;
      }
    }
    __syncthreads();
  }
}

// ---------------------------------------------------------------------------
extern "C" void kernel_launch(void* const* d_in, const int* in_sizes, int n_in,
                              void* d_out, int out_size, void* d_ws, size_t ws_size,
                              hipStream_t stream) {
  (void)in_sizes; (void)n_in; (void)out_size; (void)ws_size;
  const float* x      = (const float*)d_in[0];  // (n,5)
  const float* mult   = (const float*)d_in[1];  // (n,1)
  const float* priors = (const float*)d_in[2];  // (n,4)

  // Workspace layout (dwords); SC_PAD keeps bbx_s 16-byte aligned.
  float* ws    = (float*)d_ws;
  float* key   = ws;                         // NPAD
  int*   sidx  = (int*)(ws + NPAD);          // NPAD
  float* bbx   = ws + 2 * NPAD;              // N_BOX*4
  float* score = bbx + N_BOX * 4;            // SC_PAD
  float* bbx_s = score + SC_PAD;             // N_BOX*4 (16B aligned)
  float* sc_s  = bbx_s + N_BOX * 4;          // N_BOX

  float* out  = (float*)d_out;               // n*5 rows
  float* keep = out + N_BOX * 5;             // n flags

  ssd_transform_kernel<<<(NPAD + 255) / 256, 256, 0, stream>>>(
      x, mult, priors, key, sidx, bbx, score);
  ssd_sort_kernel<<<1, 1024, 0, stream>>>(key, sidx, bbx, score, bbx_s, sc_s);
  ssd_nms_kernel<<<1, 1024, 0, stream>>>(bbx_s, sc_s, out, keep);
}